// e_ECELoss_17343077941754
// MI455X (gfx1250) — compile-verified
//
#include <hip/hip_runtime.h>
#include <hip/hip_bf16.h>

#define NBINS 10
#define TPB   256
#define SLOT  31          // 30 used floats + 1 pad word; 31 odd -> (31t+j)%64 conflict-free banks
#define NWG   1536

struct Smem {
  float4 conf[2][TPB][2]; // double-buffered staging, 16 KB
  int4   corr[2][TPB][2]; // double-buffered staging, 16 KB
  float  hist[TPB * SLOT];// per-thread private histograms, 31 KB  (total 63 KB < 64 KB cap)
};

// ---- CDNA5 async global->LDS pipeline (ASYNCcnt-tracked) ----
__device__ __forceinline__ unsigned lds_off(const void* p) {
  // low 32 bits of the flat shared-aperture address == LDS byte offset
  return (unsigned)(unsigned long long)p;
}
__device__ __forceinline__ void async_ld_b128(unsigned dst_lds, const void* gaddr) {
  asm volatile("global_load_async_to_lds_b128 %0, %1, off"
               :: "v"(dst_lds), "v"(gaddr)
               : "memory");
}
__device__ __forceinline__ void wait_async_4() { asm volatile("s_wait_asynccnt 4" ::: "memory"); }
__device__ __forceinline__ void wait_async_0() { asm volatile("s_wait_asynccnt 0" ::: "memory"); }

__global__ __launch_bounds__(TPB)
void ece_hist_kernel(const float* __restrict__ conf,
                     const int*   __restrict__ corr,
                     float*       __restrict__ ws,
                     int n) {
  __shared__ Smem s;
  const int      tid    = threadIdx.x;
  const unsigned gid    = blockIdx.x * TPB + tid;
  const unsigned stride = gridDim.x * TPB;
  const unsigned npair  = (unsigned)n >> 3;       // pairs of float4 (N % 8 == 0)

  // zero my private histogram (only I touch it until the final barrier)
  float* myhist = &s.hist[tid * SLOT];
#pragma unroll
  for (int j = 0; j < 3 * NBINS; ++j) myhist[j] = 0.0f;

  const unsigned iters = (npair + stride - 1) / stride;

  // prologue: stage iteration 0 into buffer 0 (clamped address => uniform ASYNCcnt bookkeeping)
  {
    unsigned p0 = gid < npair ? gid : (npair - 1);
    const float* gc = conf + (size_t)p0 * 8;
    const int*   ga = corr + (size_t)p0 * 8;
    async_ld_b128(lds_off(&s.conf[0][tid][0]), gc);
    async_ld_b128(lds_off(&s.conf[0][tid][1]), gc + 4);
    async_ld_b128(lds_off(&s.corr[0][tid][0]), ga);
    async_ld_b128(lds_off(&s.corr[0][tid][1]), ga + 4);
  }

  for (unsigned it = 0; it < iters; ++it) {
    const unsigned buf = it & 1u;
    if (it + 1 < iters) {
      unsigned pn = gid + (it + 1) * stride;
      if (pn >= npair) pn = npair - 1;
      const float* gc = conf + (size_t)pn * 8;
      const int*   ga = corr + (size_t)pn * 8;
      async_ld_b128(lds_off(&s.conf[buf ^ 1][tid][0]), gc);
      async_ld_b128(lds_off(&s.conf[buf ^ 1][tid][1]), gc + 4);
      async_ld_b128(lds_off(&s.corr[buf ^ 1][tid][0]), ga);
      async_ld_b128(lds_off(&s.corr[buf ^ 1][tid][1]), ga + 4);
      wait_async_4();                             // 4 just-issued may be in flight; current buf done
    } else {
      wait_async_0();
    }

    const unsigned p = gid + it * stride;
    if (p < npair) {
#pragma unroll
      for (int h = 0; h < 2; ++h) {
        const float4 c4 = s.conf[buf][tid][h];
        const int4   a4 = s.corr[buf][tid][h];
        const float cs[4] = {c4.x, c4.y, c4.z, c4.w};
        const int   as[4] = {a4.x, a4.y, a4.z, a4.w};
#pragma unroll
        for (int k = 0; k < 4; ++k) {
          const float c = cs[k];
          int b = (int)ceilf(c * 10.0f) - 1;      // bin = ceil(10c)-1, clipped
          b = b < 0 ? 0 : (b > NBINS - 1 ? NBINS - 1 : b);
          // valid <=> 0 < c <= 1.0 <=> (bits(c)-1) <u bits(1.0); handles 0, neg, >1, NaN
          const unsigned cb = __float_as_uint(c);
          const bool  valid = (cb - 1u) < 0x3f800000u;
          const float w  = valid ? 1.0f : 0.0f;
          const float cw = valid ? c    : 0.0f;
          const float aw = valid ? (float)as[k] : 0.0f;
          float* slot = myhist + b * 3;
          atomicAdd(slot + 0, w);                 // ds_add_f32, no-return, conflict-free banks
          atomicAdd(slot + 1, cw);
          atomicAdd(slot + 2, aw);
        }
      }
    }
  }

  __syncthreads();
  // 30 lanes sweep the 256 private copies (columns j=0..29; bank = (31t+j)%64 -> conflict-free)
  if (tid < 3 * NBINS) {
    float acc = 0.0f;
    for (int t = 0; t < TPB; ++t) acc += s.hist[t * SLOT + tid];
    atomicAdd(&ws[tid], acc);                     // one global f32 atomic per quantity per WG
  }
}

__global__ void ece_finalize_kernel(const float* __restrict__ ws,
                                    float* __restrict__ out, float invN) {
  if (blockIdx.x == 0 && threadIdx.x == 0) {
    float ece = 0.0f;
#pragma unroll
    for (int b = 0; b < NBINS; ++b) {
      const float cnt = ws[b * 3 + 0];
      const float sc  = ws[b * 3 + 1];
      const float sa  = ws[b * 3 + 2];
      const float safe = fmaxf(cnt, 1.0f);
      const float gap  = fabsf(sc / safe - sa / safe);
      if (cnt > 0.0f) ece += gap * cnt * invN;
    }
    out[0] = ece;
  }
}

extern "C" void kernel_launch(void* const* d_in, const int* in_sizes, int n_in,
                              void* d_out, int out_size, void* d_ws, size_t ws_size,
                              hipStream_t stream) {
  const float* conf = (const float*)d_in[0];
  const int*   corr = (const int*)d_in[1];
  float*       out  = (float*)d_out;
  float*       ws   = (float*)d_ws;
  const int n = in_sizes[0];

  hipMemsetAsync(ws, 0, 3 * NBINS * sizeof(float), stream);   // graph-capturable
  ece_hist_kernel<<<NWG, TPB, 0, stream>>>(conf, corr, ws, n);
  ece_finalize_kernel<<<1, 32, 0, stream>>>(ws, out, 1.0f / (float)n);
}